// TopKGate_3848290697288
// MI455X (gfx1250) — compile-verified
//
#include <hip/hip_runtime.h>
#include <math.h>

typedef __attribute__((ext_vector_type(2))) float v2f;
typedef __attribute__((ext_vector_type(4))) float v4f;
typedef __attribute__((ext_vector_type(8))) float v8f;

#define S_TOKENS 8192
#define HIDDEN   4096
#define NEXP     64
#define CAP      256
#define CHUNK    256
#define NCHUNK   (S_TOKENS / CHUNK)   /* 32 */
#define TILE_K   64

// ---------------- GEMM: logits[s,e] = sum_k x[s,k]*W[e,k], fp32 WMMA 16x16x4 ----------------
__global__ __launch_bounds__(256) void moe_gemm_logits(
    const float* __restrict__ x, const float* __restrict__ W, float* __restrict__ logits) {
  __shared__ float sx[32][TILE_K + 4];
  __shared__ float sw[NEXP][TILE_K + 4];
  const int tid   = threadIdx.x;
  const int lane  = tid & 31;
  const int wave  = tid >> 5;
  const int laneM = lane & 15;
  const int koff  = (lane >> 4) << 1;      // 0 or 2 (A/B K-pair base per lane half)
  const int m0l   = (wave >> 2) * 16;      // local row tile: 0 or 16
  const int n0    = (wave & 3) * 16;       // expert tile: 0,16,32,48
  const int blockRow = blockIdx.x * 32;

  v8f acc = {};

  for (int kb = 0; kb < HIDDEN; kb += TILE_K) {
    __syncthreads();
    // x tile: 32 rows x 64 K  (512 x 16B, 2 per thread)
    for (int i = tid; i < 32 * (TILE_K / 4); i += 256) {
      const int r = i >> 4;
      const int c = (i & 15) << 2;
      const v4f v = *(const v4f*)(x + (size_t)(blockRow + r) * HIDDEN + kb + c);
      *(v4f*)&sx[r][c] = v;
    }
    // W tile: 64 experts x 64 K (1024 x 16B, 4 per thread)
    for (int i = tid; i < NEXP * (TILE_K / 4); i += 256) {
      const int r = i >> 4;
      const int c = (i & 15) << 2;
      const v4f v = *(const v4f*)(W + (size_t)r * HIDDEN + kb + c);
      *(v4f*)&sw[r][c] = v;
    }
    __syncthreads();
    #pragma unroll
    for (int kk = 0; kk < TILE_K; kk += 4) {
      v2f a, b;
      a.x = sx[m0l + laneM][kk + koff];
      a.y = sx[m0l + laneM][kk + koff + 1];
      b.x = sw[n0 + laneM][kk + koff];        // B[k][n] = W[n][k]
      b.y = sw[n0 + laneM][kk + koff + 1];
      acc = __builtin_amdgcn_wmma_f32_16x16x4_f32(false, a, false, b,
                                                  (short)0, acc, false, false);
    }
  }
  // C layout: VGPR i -> (M=i, N=lane) lanes 0-15 ; (M=8+i, N=lane-16) lanes 16-31
  const int n = n0 + laneM;
  const int mbase = blockRow + m0l + ((lane >> 4) << 3);
  #pragma unroll
  for (int i = 0; i < 8; ++i)
    logits[(size_t)(mbase + i) * NEXP + n] = acc[i];
}

// ---------------- softmax + top2 + chunk histograms + deterministic column sums ----------------
__global__ __launch_bounds__(256) void moe_gate(
    const float* __restrict__ logits,
    int* __restrict__ sel1, int* __restrict__ sel2,
    float* __restrict__ g1o, float* __restrict__ g2o,
    int* __restrict__ r1o, int* __restrict__ r2o,
    int* __restrict__ cnt1g, int* __restrict__ cnt2g,
    float* __restrict__ gsumg) {
  __shared__ int   ls1[CHUNK], ls2[CHUNK];
  __shared__ int   c1[NEXP], c2[NEXP];
  __shared__ float wsum[8][NEXP];
  const int t = threadIdx.x;
  const int chunk = blockIdx.x;
  const int s = chunk * CHUNK + t;
  const int wave = t >> 5, lane = t & 31;

  if (t < NEXP) { c1[t] = 0; c2[t] = 0; }
  __syncthreads();

  const float* row = logits + (size_t)s * NEXP;
  float v1 = -INFINITY, v2 = -INFINITY;
  int i1 = 0, i2 = 0;
  for (int e = 0; e < NEXP; ++e) {           // ties -> lowest index, matches top_k
    const float l = row[e];
    if (l > v1) { v2 = v1; i2 = i1; v1 = l; i1 = e; }
    else if (l > v2) { v2 = l; i2 = e; }
  }
  const float m = v1;
  float sum = 0.f;
  for (int e = 0; e < NEXP; ++e) sum += expf(row[e] - m);
  const float inv = 1.f / sum;
  const float g1 = inv;                      // exp(v1-m)=1
  const float g2 = expf(v2 - m) * inv;

  ls1[t] = i1; ls2[t] = i2;
  atomicAdd(&c1[i1], 1);                     // integer atomics: order-independent
  atomicAdd(&c2[i2], 1);

  // deterministic per-wave tree reduce of gate columns (for me[e])
  for (int e = 0; e < NEXP; ++e) {
    float g = expf(row[e] - m) * inv;
    for (int off = 16; off > 0; off >>= 1) g += __shfl_xor(g, off, 32);
    if (lane == 0) wsum[wave][e] = g;
  }
  __syncthreads();

  // in-chunk rank: tokens before me in this chunk with same expert (exact cumsum order)
  int r1 = 0, r2 = 0;
  for (int j = 0; j < t; ++j) {
    r1 += (ls1[j] == i1);
    r2 += (ls2[j] == i2);
  }
  sel1[s] = i1; sel2[s] = i2;
  g1o[s] = g1;  g2o[s] = g2;
  r1o[s] = r1;  r2o[s] = r2;
  if (t < NEXP) {
    cnt1g[chunk * NEXP + t] = c1[t];
    cnt2g[chunk * NEXP + t] = c2[t];
    float tot = 0.f;
    for (int w = 0; w < 8; ++w) tot += wsum[w][t];   // fixed order
    gsumg[chunk * NEXP + t] = tot;
  }
}

// ---------------- per-expert exclusive scan over chunks + l_aux + exp_counts ----------------
__global__ void moe_scan(const int* __restrict__ cnt1g, const int* __restrict__ cnt2g,
                         const float* __restrict__ gsumg,
                         int* __restrict__ base1, int* __restrict__ base2,
                         int* __restrict__ total1,
                         float* __restrict__ laux_out, float* __restrict__ expcnt_out) {
  __shared__ float prod[NEXP];
  const int e = threadIdx.x;                 // blockDim == 64
  int run1 = 0;
  for (int c = 0; c < NCHUNK; ++c) { base1[c * NEXP + e] = run1; run1 += cnt1g[c * NEXP + e]; }
  total1[e] = run1;                          // sum(mask1) pre-drop
  expcnt_out[e] = (float)run1;               // exp_counts
  int run2 = 0;
  for (int c = 0; c < NCHUNK; ++c) { base2[c * NEXP + e] = run2; run2 += cnt2g[c * NEXP + e]; }
  float gs = 0.f;
  for (int c = 0; c < NCHUNK; ++c) gs += gsumg[c * NEXP + e];
  const float me = gs * (1.f / S_TOKENS);
  const float ce = (float)run1 * (1.f / S_TOKENS);
  prod[e] = me * ce;
  __syncthreads();
  if (e == 0) {
    float a = 0.f;
    for (int i = 0; i < NEXP; ++i) a += prod[i];   // fixed order
    *laux_out = a * (float)NEXP;             // mean(me*ce)*E*E == sum*E
  }
}

// ---------------- capacity drop, renormalize, sparse scatter ----------------
__global__ __launch_bounds__(256) void moe_scatter(
    const int* __restrict__ sel1, const int* __restrict__ sel2,
    const float* __restrict__ g1i, const float* __restrict__ g2i,
    const int* __restrict__ r1i, const int* __restrict__ r2i,
    const int* __restrict__ base1, const int* __restrict__ base2,
    const int* __restrict__ total1,
    float* __restrict__ combine, float* __restrict__ dispatch) {
  const int t = threadIdx.x, chunk = blockIdx.x;
  const int s = chunk * CHUNK + t;
  const int e1 = sel1[s], e2 = sel2[s];
  const int loc1 = base1[chunk * NEXP + e1] + r1i[s];
  const int loc2 = total1[e2] + base2[chunk * NEXP + e2] + r2i[s];  // + sum(mask1) offset
  const bool k1 = loc1 < CAP, k2 = loc2 < CAP;
  const float g1 = k1 ? g1i[s] : 0.f;        // drop BEFORE renormalization
  const float g2 = k2 ? g2i[s] : 0.f;
  const float denom = fmaxf(g1 + g2, 1.1920928955078125e-07f);  // float32 eps
  const float w1 = g1 / denom, w2 = g2 / denom;
  if (k1) {
    const size_t off = ((size_t)s * NEXP + e1) * CAP + loc1;
    __builtin_nontemporal_store(w1, &combine[off]);
    __builtin_nontemporal_store((w1 != 0.f) ? 1.f : 0.f, &dispatch[off]);
  }
  if (k2) {
    const size_t off = ((size_t)s * NEXP + e2) * CAP + loc2;
    __builtin_nontemporal_store(w2, &combine[off]);
    __builtin_nontemporal_store((w2 != 0.f) ? 1.f : 0.f, &dispatch[off]);
  }
}

// ---------------- streaming zero-fill of d_out (NT stores, 1.07 GB) ----------------
__global__ void zero_out(v4f* __restrict__ p, long long n4,
                         float* __restrict__ tail, int ntail) {
  const long long i0 = (long long)blockIdx.x * blockDim.x + threadIdx.x;
  const long long stride = (long long)gridDim.x * blockDim.x;
  const v4f z = {0.f, 0.f, 0.f, 0.f};
  for (long long i = i0; i < n4; i += stride) __builtin_nontemporal_store(z, &p[i]);
  if (i0 == 0) for (int i = 0; i < ntail; ++i) tail[i] = 0.f;
}

extern "C" void kernel_launch(void* const* d_in, const int* in_sizes, int n_in,
                              void* d_out, int out_size, void* d_ws, size_t ws_size,
                              hipStream_t stream) {
  (void)in_sizes; (void)n_in; (void)out_size; (void)ws_size;
  const float* x = (const float*)d_in[0];
  const float* W = (const float*)d_in[1];
  float* out = (float*)d_out;

  // workspace layout (~2.3 MB)
  float* logits = (float*)d_ws;
  char* p = (char*)d_ws + (size_t)S_TOKENS * NEXP * sizeof(float);
  int*   sel1  = (int*)p;   p += S_TOKENS * 4;
  int*   sel2  = (int*)p;   p += S_TOKENS * 4;
  float* g1    = (float*)p; p += S_TOKENS * 4;
  float* g2    = (float*)p; p += S_TOKENS * 4;
  int*   r1    = (int*)p;   p += S_TOKENS * 4;
  int*   r2    = (int*)p;   p += S_TOKENS * 4;
  int*   cnt1g = (int*)p;   p += NCHUNK * NEXP * 4;
  int*   cnt2g = (int*)p;   p += NCHUNK * NEXP * 4;
  int*   base1 = (int*)p;   p += NCHUNK * NEXP * 4;
  int*   base2 = (int*)p;   p += NCHUNK * NEXP * 4;
  int*   total1= (int*)p;   p += NEXP * 4;
  float* gsumg = (float*)p; p += NCHUNK * NEXP * 4;

  const size_t SEC = (size_t)S_TOKENS * NEXP * CAP;   // 134217728
  float* laux     = out;
  float* combine  = out + 1;
  float* dispatch = out + 1 + SEC;
  float* expcnt   = out + 1 + 2 * SEC;

  const long long total = 1 + 2 * (long long)SEC + NEXP;  // 268435521
  const long long n4 = total / 4;
  const int ntail = (int)(total - n4 * 4);

  zero_out<<<4096, 256, 0, stream>>>((v4f*)d_out, n4, (float*)d_out + n4 * 4, ntail);
  moe_gemm_logits<<<S_TOKENS / 32, 256, 0, stream>>>(x, W, logits);
  moe_gate<<<NCHUNK, 256, 0, stream>>>(logits, sel1, sel2, g1, g2, r1, r2,
                                       cnt1g, cnt2g, gsumg);
  moe_scan<<<1, 64, 0, stream>>>(cnt1g, cnt2g, gsumg, base1, base2, total1, laux, expcnt);
  moe_scatter<<<NCHUNK, 256, 0, stream>>>(sel1, sel2, g1, g2, r1, r2,
                                          base1, base2, total1, combine, dispatch);
}